// SpatialTransformer_52810917872172
// MI455X (gfx1250) — compile-verified
//
#include <hip/hip_runtime.h>

// Problem shape (from setup_inputs): img [4,2,128,160,160] f32, flow [4,3,128,160,160] f32
#define B_   4
#define C_   2
#define D_   128
#define H_   160
#define W_   160
#define DHW_ (D_ * H_ * W_)            // 3,276,800
#define BLKX_ 80                       // x-pairs per row (W/2)
#define BLKY_ 2                        // rows per block
#define NT_   (BLKX_ * BLKY_)          // 160 threads = 5 waves

typedef float v2f __attribute__((ext_vector_type(2)));
typedef int   v2i __attribute__((ext_vector_type(2)));

typedef __attribute__((address_space(1))) v2i as1_v2i;
typedef __attribute__((address_space(3))) v2i as3_v2i;

// Faithful port of the module's normalization + grid_sample unnormalization:
// c = 2*((g+f)/(S-1) - 0.5);  p = ((c+1)*S - 1)/2
__device__ __forceinline__ float samplepos(float g, float f, float S) {
    float c = 2.0f * ((g + f) / (S - 1.0f) - 0.5f);
    return ((c + 1.0f) * S - 1.0f) * 0.5f;
}

__device__ __forceinline__ int iclamp(int v, int lo, int hi) {
    return v < lo ? lo : (v > hi ? hi : v);
}

#if defined(__gfx1250__) && __has_builtin(__builtin_amdgcn_global_load_async_to_lds_b64)
#define ASYNC_FLOW 1
#else
#define ASYNC_FLOW 0
#endif

// Launch: grid = (1, H/BLKY_, B*D), block = (BLKX_, BLKY_)
__global__ __launch_bounds__(NT_) void warp3d_kernel(const float* __restrict__ img,
                                                     const float* __restrict__ flow,
                                                     float* __restrict__ out)
{
    // Uniform (per-block, SALU) coordinates: no integer division anywhere.
    const int bz = blockIdx.z;          // 0 .. B*D-1
    const int z  = bz & (D_ - 1);       // D is a power of two
    const int b  = bz >> 7;             // bz / D
    const int y  = blockIdx.y * BLKY_ + threadIdx.y;
    const int x0e = threadIdx.x * 2;    // even x of this thread's voxel pair
    const int lidx = threadIdx.y * BLKX_ + threadIdx.x;

    const int zyx = (z * H_ + y) * W_ + x0e;
    const float* fb = flow + (size_t)b * (3 * DHW_) + zyx;

    // ---- Stage the 6 flow values (3 axes x 2 voxels) -----------------------
    float fz[2], fy[2], fx[2];
#if ASYNC_FLOW
    // gfx1250 async global->LDS engine (ASYNCcnt). Each lane owns its own
    // 8-byte LDS slot, so only s_wait_asynccnt is needed (no barrier).
    __shared__ __align__(16) float sflow[3][2 * NT_];
    {
        #pragma unroll
        for (int a = 0; a < 3; ++a) {
            __builtin_amdgcn_global_load_async_to_lds_b64(
                (as1_v2i*)(const void*)(fb + a * DHW_),
                (as3_v2i*)(void*)&sflow[a][lidx * 2],
                /*offset=*/0, /*cpol=*/0);
        }
#if __has_builtin(__builtin_amdgcn_s_wait_asynccnt)
        __builtin_amdgcn_s_wait_asynccnt(0);
#else
        asm volatile("s_wait_asynccnt 0" ::: "memory");
#endif
        asm volatile("" ::: "memory");  // compiler barrier: keep LDS reads below the wait
        fz[0] = sflow[0][lidx * 2]; fz[1] = sflow[0][lidx * 2 + 1];
        fy[0] = sflow[1][lidx * 2]; fy[1] = sflow[1][lidx * 2 + 1];
        fx[0] = sflow[2][lidx * 2]; fx[1] = sflow[2][lidx * 2 + 1];
    }
#else
    {
        // Streamed exactly once -> non-temporal so L2 stays reserved for img.
        v2f f0 = __builtin_nontemporal_load((const v2f*)(fb + 0 * DHW_));
        v2f f1 = __builtin_nontemporal_load((const v2f*)(fb + 1 * DHW_));
        v2f f2 = __builtin_nontemporal_load((const v2f*)(fb + 2 * DHW_));
        fz[0] = f0.x; fz[1] = f0.y;
        fy[0] = f1.x; fy[1] = f1.y;
        fx[0] = f2.x; fx[1] = f2.y;
    }
#endif

    const float* imgb = img + (size_t)b * (C_ * DHW_);
    float res[C_][2];

    #pragma unroll
    for (int v = 0; v < 2; ++v) {
        const float pz = samplepos((float)z,         fz[v], (float)D_);
        const float py = samplepos((float)y,         fy[v], (float)H_);
        const float px = samplepos((float)(x0e + v), fx[v], (float)W_);

        const float zf = floorf(pz), yf = floorf(py), xf = floorf(px);
        const float tz = pz - zf,    ty = py - yf,    tx = px - xf;
        const int z0 = (int)zf, y0 = (int)yf, x0 = (int)xf;
        const int z1 = z0 + 1,  y1 = y0 + 1,  x1 = x0 + 1;

        // Branchless zeros-padding: invalid corner => zero weight on that axis.
        const float wz0 = (z0 >= 0 && z0 < D_) ? (1.0f - tz) : 0.0f;
        const float wz1 = (z1 >= 0 && z1 < D_) ? tz          : 0.0f;
        const float wy0 = (y0 >= 0 && y0 < H_) ? (1.0f - ty) : 0.0f;
        const float wy1 = (y1 >= 0 && y1 < H_) ? ty          : 0.0f;
        const float wx0 = (x0 >= 0 && x0 < W_) ? (1.0f - tx) : 0.0f;
        const float wx1 = (x1 >= 0 && x1 < W_) ? tx          : 0.0f;

        const int cz0 = iclamp(z0, 0, D_ - 1), cz1 = iclamp(z1, 0, D_ - 1);
        const int cy0 = iclamp(y0, 0, H_ - 1), cy1 = iclamp(y1, 0, H_ - 1);
        const int cx0 = iclamp(x0, 0, W_ - 1), cx1 = iclamp(x1, 0, W_ - 1);

        const int r00 = (cz0 * H_ + cy0) * W_;
        const int r01 = (cz0 * H_ + cy1) * W_;
        const int r10 = (cz1 * H_ + cy0) * W_;
        const int r11 = (cz1 * H_ + cy1) * W_;

        #pragma unroll
        for (int c = 0; c < C_; ++c) {
            const float* pc = imgb + c * DHW_;  // gathers hit L2-resident img
            res[c][v] =
                wz0 * (wy0 * (wx0 * pc[r00 + cx0] + wx1 * pc[r00 + cx1]) +
                       wy1 * (wx0 * pc[r01 + cx0] + wx1 * pc[r01 + cx1])) +
                wz1 * (wy0 * (wx0 * pc[r10 + cx0] + wx1 * pc[r10 + cx1]) +
                       wy1 * (wx0 * pc[r11 + cx0] + wx1 * pc[r11 + cx1]));
        }
    }

    // Written once, never re-read -> non-temporal b64 stores (don't pollute L2).
    float* ob = out + (size_t)b * (C_ * DHW_) + zyx;
    v2f s0 = { res[0][0], res[0][1] };
    v2f s1 = { res[1][0], res[1][1] };
    __builtin_nontemporal_store(s0, (v2f*)ob);
    __builtin_nontemporal_store(s1, (v2f*)(ob + DHW_));
}

extern "C" void kernel_launch(void* const* d_in, const int* in_sizes, int n_in,
                              void* d_out, int out_size, void* d_ws, size_t ws_size,
                              hipStream_t stream) {
    (void)in_sizes; (void)n_in; (void)out_size; (void)d_ws; (void)ws_size;
    const float* img  = (const float*)d_in[0];
    const float* flow = (const float*)d_in[1];
    float* out        = (float*)d_out;

    dim3 block(BLKX_, BLKY_, 1);        // 160 threads = 5 wave32
    dim3 grid(1, H_ / BLKY_, B_ * D_);  // (1, 80, 512)
    warp3d_kernel<<<grid, block, 0, stream>>>(img, flow, out);
}